// GRAPH_32985348833663
// MI455X (gfx1250) — compile-verified
//
#include <hip/hip_runtime.h>
#include <math.h>

// ---------- constants ----------
#define NUM_USERC 20000
#define NUM_ITEMC 80000
#define NNODES    100000
#define DIM       128
#define DIMX      64
#define NEDGE     800000
#define NBATCH    4096
#define SLOPE     0.01f

typedef float v2f __attribute__((ext_vector_type(2)));
typedef float v8f __attribute__((ext_vector_type(8)));

__device__ __forceinline__ float lrelu(float v) { return v >= 0.f ? v : SLOPE * v; }

__device__ __forceinline__ float wave_sum(float v) {
    #pragma unroll
    for (int off = 16; off > 0; off >>= 1) v += __shfl_xor(v, off, 32);
    return v;
}

// async global -> LDS copy of 16 bytes (CDNA5 GLOBAL_LOAD_ASYNC_TO_LDS_B128)
__device__ __forceinline__ void async_copy16(void* lds_dst, const float* gsrc) {
    unsigned int       l = (unsigned int)(unsigned long long)lds_dst;  // low 32b = LDS offset
    unsigned long long g = (unsigned long long)gsrc;
    asm volatile("global_load_async_to_lds_b128 %0, %1, off" :: "v"(l), "v"(g) : "memory");
}
__device__ __forceinline__ void async_wait0() {
    asm volatile("s_wait_asynccnt 0x0" ::: "memory");
}

// ---------- utility kernels ----------
__global__ void k_zero(float* __restrict__ p, long long n) {
    long long i = (long long)blockIdx.x * blockDim.x + threadIdx.x;
    if (i < n) p[i] = 0.f;
}

// Pack a weight matrix into WMMA B-fragment order:
// P[((ct*32+s)*32+L)*2 + {0,1}] = {B[k][n], B[k+1][n]},
//   n = ct*16 + (L&15), k = 4*s + 2*(L>>4)
// transposed=1: B[k][n] = W[n*K+k]  (for X @ W.T)
// transposed=0: B[k][n] = W[k*OUTD+n]
__global__ void k_pack(const float* __restrict__ W, float* __restrict__ P,
                       int OUTD, int K, int transposed) {
    int i = blockIdx.x * blockDim.x + threadIdx.x;
    int total = (OUTD >> 4) << 10;           // coltiles * 32 steps * 32 lanes
    if (i >= total) return;
    int L = i & 31, s = (i >> 5) & 31, ct = i >> 10;
    int n = ct * 16 + (L & 15);
    int k = 4 * s + ((L >> 4) << 1);
    float b0, b1;
    if (transposed) { b0 = W[n * K + k];    b1 = W[n * K + k + 1]; }
    else            { b0 = W[k * OUTD + n]; b1 = W[(k + 1) * OUTD + n]; }
    P[2 * i] = b0; P[2 * i + 1] = b1;
}

// ---------- embed: x = l2norm(tanh(in @ W.T + b)) ----------
// block = 256 (8 waves), 16-row tile, wave w owns cols [16w,16w+16)
__global__ __launch_bounds__(256) void k_embed(
    const float* __restrict__ feat, const float* __restrict__ ufeat,
    const float* __restrict__ PF, const float* __restrict__ bF,
    const float* __restrict__ PU, const float* __restrict__ bU,
    float* __restrict__ x)
{
    __shared__ float As[16][132];
    __shared__ float Os[16][132];
    __shared__ float nrm[16];
    const int t = threadIdx.x, lane = t & 31, wave = t >> 5;
    const int r0 = blockIdx.x * 16;
    const bool isUser = (r0 >= NUM_ITEMC);   // 80000 % 16 == 0: tiles never straddle
    const float* Arows = isUser ? (ufeat + (long long)(r0 - NUM_ITEMC) * DIM)
                                : (feat + (long long)r0 * DIM);
    const float* Wp   = isUser ? PU : PF;
    const float* bias = isUser ? bU : bF;

    for (int q = t; q < 512; q += 256) {            // 16 rows x 32 float4 chunks
        int rr = q >> 5, c4 = (q & 31) << 2;
        async_copy16(&As[rr][c4], Arows + rr * DIM + c4);
    }
    async_wait0();
    if (t < 16) nrm[t] = 0.f;
    __syncthreads();

    const int m = lane & 15;
    const int koff = (lane >> 4) << 1;
    const int n = wave * 16 + (lane & 15);
    const v2f* Bp = (const v2f*)Wp + (wave << 10) + lane;   // fragment-ordered B
    v8f c = {};
    #pragma unroll
    for (int s = 0; s < 32; ++s) {
        v2f a, b;
        a.x = As[m][4 * s + koff]; a.y = As[m][4 * s + koff + 1];
        b = Bp[s << 5];
        c = __builtin_amdgcn_wmma_f32_16x16x4_f32(false, a, false, b, (short)0, c, false, false);
    }
    const float bb = bias[n];
    const int mb = (lane >> 4) << 3;                // rows r / r+8 per C layout
    #pragma unroll
    for (int r = 0; r < 8; ++r) Os[r + mb][n] = tanhf(c[r] + bb);
    __syncthreads();

    {   // row sum of squares: 16 threads per row, 8 cols each
        int row = t & 15, seg = t >> 4;
        float s = 0.f;
        #pragma unroll
        for (int j = 0; j < 8; ++j) { float v = Os[row][seg * 8 + j]; s += v * v; }
        atomicAdd(&nrm[row], s);
    }
    __syncthreads();
    {   // normalized store, coalesced
        int row = t >> 4, c0 = t & 15;
        float sc = 1.f / fmaxf(sqrtf(nrm[row]), 1e-12f);
        #pragma unroll
        for (int j = 0; j < 8; ++j) {
            int col = c0 + 16 * j;
            x[(long long)(r0 + row) * DIM + col] = Os[row][col] * sc;
        }
    }
}

// ---------- xw = x @ W_gat (packed B fragments) ----------
__global__ __launch_bounds__(256) void k_gemm_xw(
    const float* __restrict__ x, const float* __restrict__ Pgat, float* __restrict__ xw)
{
    __shared__ float As[16][132];
    const int t = threadIdx.x, lane = t & 31, wave = t >> 5;
    const int r0 = blockIdx.x * 16;
    for (int q = t; q < 512; q += 256) {
        int rr = q >> 5, c4 = (q & 31) << 2;
        async_copy16(&As[rr][c4], x + (long long)(r0 + rr) * DIM + c4);
    }
    async_wait0();
    __syncthreads();
    const int m = lane & 15, koff = (lane >> 4) << 1, n = wave * 16 + (lane & 15);
    const v2f* Bp = (const v2f*)Pgat + (wave << 10) + lane;
    v8f c = {};
    #pragma unroll
    for (int s = 0; s < 32; ++s) {
        v2f a, b;
        a.x = As[m][4 * s + koff]; a.y = As[m][4 * s + koff + 1];
        b = Bp[s << 5];
        c = __builtin_amdgcn_wmma_f32_16x16x4_f32(false, a, false, b, (short)0, c, false, false);
    }
    const int mb = (lane >> 4) << 3;
    #pragma unroll
    for (int r = 0; r < 8; ++r)
        xw[(long long)(r0 + r + mb) * DIM + n] = c[r];
}

// ---------- deg = segment_sum(ones, src) ----------
__global__ void k_deg(const int* __restrict__ src, float* __restrict__ deg) {
    int e = blockIdx.x * blockDim.x + threadIdx.x;
    if (e < NEDGE) atomicAdd(&deg[src[e]], 1.f);
}

// ---------- per-edge logits + segment max (order-preserving uint atomicMax) ----------
__global__ __launch_bounds__(256) void k_edge_logits(
    const int* __restrict__ src, const int* __restrict__ dst,
    const float* __restrict__ xw, const float* __restrict__ deg,
    float* __restrict__ logits, unsigned int* __restrict__ mx)
{
    const int lane = threadIdx.x & 31, wave = threadIdx.x >> 5;
    const int e = blockIdx.x * 8 + wave;
    if (e >= NEDGE) return;
    const int s = src[e], d = dst[e];
    const float4 vj = *(const float4*)(xw + (long long)s * DIM + lane * 4);
    const float4 vi = *(const float4*)(xw + (long long)d * DIM + lane * 4);
    float p = vj.x * vi.x + vj.y * vi.y + vj.z * vi.z + vj.w * vi.w;
    p = wave_sum(p);
    if (lane == 0) {
        float dinv = 1.f / sqrtf(deg[s]);
        float gate = 1.f / (1.f + expf(-dinv * p));
        float lg = p * gate;
        logits[e] = lg;
        unsigned int u = __float_as_uint(lg);
        u = (u & 0x80000000u) ? ~u : (u | 0x80000000u);
        atomicMax(&mx[d], u);
    }
}

// ---------- decode segment-max keys back to float (in place) ----------
__global__ void k_decode_max(unsigned int* __restrict__ mx, int n) {
    int i = blockIdx.x * blockDim.x + threadIdx.x;
    if (i < n) {
        unsigned int u = mx[i];
        mx[i] = (u & 0x80000000u) ? (u & 0x7fffffffu) : ~u;
    }
}

// ---------- e = exp(logits - m[dst]); sm[dst] += e ----------
__global__ void k_expsum(const int* __restrict__ dst, float* __restrict__ logits,
                         const float* __restrict__ mxf, float* __restrict__ sm) {
    int e = blockIdx.x * blockDim.x + threadIdx.x;
    if (e < NEDGE) {
        int d = dst[e];
        float ev = expf(logits[e] - mxf[d]);
        logits[e] = ev;
        atomicAdd(&sm[d], ev);
    }
}

// ---------- agg[dst] += xw[src] * att (wave per edge, f32 L2 atomics) ----------
__global__ __launch_bounds__(256) void k_scatter(
    const int* __restrict__ src, const int* __restrict__ dst,
    const float* __restrict__ xw, const float* __restrict__ ev,
    const float* __restrict__ sm, float* __restrict__ agg)
{
    const int lane = threadIdx.x & 31, wave = threadIdx.x >> 5;
    const int e = blockIdx.x * 8 + wave;
    if (e >= NEDGE) return;
    const int s = src[e], d = dst[e];
    float w = 0.f;
    if (lane == 0) w = ev[e] / (sm[d] + 1e-16f);
    w = __shfl(w, 0, 32);
    const float4 vj = *(const float4*)(xw + (long long)s * DIM + lane * 4);
    float* a = agg + (long long)d * DIM + lane * 4;
    atomicAdd(a + 0, vj.x * w);
    atomicAdd(a + 1, vj.y * w);
    atomicAdd(a + 2, vj.z * w);
    atomicAdd(a + 3, vj.w * w);
}

// ---------- h = lrelu(l2norm(agg + b_gat)), in place ----------
__global__ __launch_bounds__(256) void k_hnorm(float* __restrict__ agg,
                                               const float* __restrict__ bgat) {
    const int lane = threadIdx.x & 31, wave = threadIdx.x >> 5;
    const int row = blockIdx.x * 8 + wave;
    if (row >= NNODES) return;
    float* p = agg + (long long)row * DIM;
    float4 v = *(const float4*)(p + lane * 4);
    const float4 b = *(const float4*)(bgat + lane * 4);
    v.x += b.x; v.y += b.y; v.z += b.z; v.w += b.w;
    float s = v.x * v.x + v.y * v.y + v.z * v.z + v.w * v.w;
    s = wave_sum(s);
    float sc = 1.f / fmaxf(sqrtf(s), 1e-12f);
    v.x = lrelu(v.x * sc); v.y = lrelu(v.y * sc);
    v.z = lrelu(v.z * sc); v.w = lrelu(v.w * sc);
    *(float4*)(p + lane * 4) = v;
}

// ---------- rep = lrelu(h@Wg1.T + bg1 + lrelu(x@Wl1.T + bl1) + id_emb) ----------
// block = 128 (4 waves), 16-row tile, wave w owns cols [16w,16w+16) of 64
__global__ __launch_bounds__(128) void k_rep(
    const float* __restrict__ x, const float* __restrict__ h,
    const float* __restrict__ Pl1, const float* __restrict__ bl1,
    const float* __restrict__ Pg1, const float* __restrict__ bg1,
    const float* __restrict__ id_emb, float* __restrict__ rep)
{
    __shared__ float Xs[16][132];
    __shared__ float Hs[16][132];
    const int t = threadIdx.x, lane = t & 31, wave = t >> 5;
    const int r0 = blockIdx.x * 16;
    for (int q = t; q < 512; q += 128) {
        int rr = q >> 5, c4 = (q & 31) << 2;
        async_copy16(&Xs[rr][c4], x + (long long)(r0 + rr) * DIM + c4);
        async_copy16(&Hs[rr][c4], h + (long long)(r0 + rr) * DIM + c4);
    }
    async_wait0();
    __syncthreads();
    const int m = lane & 15, koff = (lane >> 4) << 1, n = wave * 16 + (lane & 15);
    const v2f* Bp1 = (const v2f*)Pl1 + (wave << 10) + lane;
    const v2f* Bp2 = (const v2f*)Pg1 + (wave << 10) + lane;
    v8f c1 = {}, c2 = {};
    #pragma unroll
    for (int s = 0; s < 32; ++s) {
        v2f a1, a2, b1, b2;
        a1.x = Xs[m][4 * s + koff]; a1.y = Xs[m][4 * s + koff + 1];
        a2.x = Hs[m][4 * s + koff]; a2.y = Hs[m][4 * s + koff + 1];
        b1 = Bp1[s << 5];
        b2 = Bp2[s << 5];
        c1 = __builtin_amdgcn_wmma_f32_16x16x4_f32(false, a1, false, b1, (short)0, c1, false, false);
        c2 = __builtin_amdgcn_wmma_f32_16x16x4_f32(false, a2, false, b2, (short)0, c2, false, false);
    }
    const float B1 = bl1[n], B2 = bg1[n];
    const int mb = (lane >> 4) << 3;
    #pragma unroll
    for (int r = 0; r < 8; ++r) {
        long long row = r0 + r + mb;
        float xh = lrelu(c1[r] + B1) + id_emb[row * DIMX + n];
        rep[row * DIMX + n] = lrelu(c2[r] + B2 + xh);
    }
}

// ---------- batch: scores + price MLP (wave per sample) ----------
__global__ __launch_bounds__(256) void k_batch(
    const float* __restrict__ rep, const int* __restrict__ un,
    const int* __restrict__ pi, const int* __restrict__ ni,
    const float* __restrict__ Wp1, const float* __restrict__ bp1,
    const float* __restrict__ Wp2, const float* __restrict__ bp2,
    float* __restrict__ pos, float* __restrict__ neg, float* __restrict__ pred)
{
    __shared__ float up[8][128];
    __shared__ float hid[8][64];
    const int lane = threadIdx.x & 31, wave = threadIdx.x >> 5;
    const int b = blockIdx.x * 8 + wave;    // NBATCH = 512*8 exactly
    const float* ur = rep + (long long)un[b] * DIMX;
    const float* pr = rep + (long long)pi[b] * DIMX;
    const float* nr = rep + (long long)ni[b] * DIMX;
    float2 u  = *(const float2*)(ur + lane * 2);
    float2 p  = *(const float2*)(pr + lane * 2);
    float2 nn = *(const float2*)(nr + lane * 2);
    float ps = wave_sum(u.x * p.x + u.y * p.y);
    float ns = wave_sum(u.x * nn.x + u.y * nn.y);
    if (lane == 0) { pos[b] = ps; neg[b] = ns; }
    up[wave][lane * 2 + 0]  = u.x; up[wave][lane * 2 + 1]  = u.y;
    up[wave][64 + lane * 2] = p.x; up[wave][65 + lane * 2] = p.y;
    __syncthreads();
    #pragma unroll
    for (int dd = 0; dd < 2; ++dd) {
        int d = lane + 32 * dd;
        float acc = bp1[d];
        for (int k = 0; k < 128; ++k) acc += up[wave][k] * Wp1[d * 128 + k];
        hid[wave][d] = lrelu(acc);
    }
    __syncthreads();
    float t2 = hid[wave][lane * 2] * Wp2[lane * 2] + hid[wave][lane * 2 + 1] * Wp2[lane * 2 + 1];
    t2 = wave_sum(t2);
    if (lane == 0) pred[b] = 1.f / (1.f + expf(-(t2 + bp2[0])));
}

// ---------- launcher ----------
extern "C" void kernel_launch(void* const* d_in, const int* in_sizes, int n_in,
                              void* d_out, int out_size, void* d_ws, size_t ws_size,
                              hipStream_t stream) {
    const float* features      = (const float*)d_in[0];
    const float* user_features = (const float*)d_in[1];
    const float* W_mlp  = (const float*)d_in[2];
    const float* b_mlp  = (const float*)d_in[3];
    const float* W_umlp = (const float*)d_in[4];
    const float* b_umlp = (const float*)d_in[5];
    const float* W_gat  = (const float*)d_in[6];
    const float* b_gat  = (const float*)d_in[7];
    const float* W_l1   = (const float*)d_in[8];
    const float* b_l1   = (const float*)d_in[9];
    const float* W_g1   = (const float*)d_in[10];
    const float* b_g1   = (const float*)d_in[11];
    const float* id_emb = (const float*)d_in[12];
    const float* Wp1    = (const float*)d_in[13];
    const float* bp1    = (const float*)d_in[14];
    const float* Wp2    = (const float*)d_in[15];
    const float* bp2    = (const float*)d_in[16];
    const int*   eidx   = (const int*)d_in[17];
    const int*   src    = eidx;
    const int*   dst    = eidx + NEDGE;
    const int*   un     = (const int*)d_in[18];
    const int*   pi     = (const int*)d_in[19];
    const int*   ni     = (const int*)d_in[20];

    float* ws = (float*)d_ws;
    size_t off = 0;
    float* x      = ws + off; off += (size_t)NNODES * DIM;
    float* xw     = ws + off; off += (size_t)NNODES * DIM;
    float* agg    = ws + off; off += (size_t)NNODES * DIM;   // becomes h in place
    float* logits = ws + off; off += (size_t)NEDGE;          // becomes exp(e) in place
    float* deg    = ws + off; off += (size_t)NNODES;
    float* sm     = ws + off; off += (size_t)NNODES;
    unsigned int* mx = (unsigned int*)(ws + off); off += (size_t)NNODES;
    float* PmlP   = ws + off; off += 128 * 128;   // fragment-packed weights
    float* PumlP  = ws + off; off += 128 * 128;
    float* PgatP  = ws + off; off += 128 * 128;
    float* Pl1    = ws + off; off += 128 * 64;
    float* Pg1    = ws + off; off += 128 * 64;

    float* outp = (float*)d_out;
    float* pos  = outp;
    float* neg  = outp + NBATCH;
    float* rep  = outp + 2 * NBATCH;
    float* pred = outp + 2 * NBATCH + (size_t)NNODES * DIMX;

    // zero accumulators (agg, deg, sm, mx)
    k_zero<<<((long long)NNODES * DIM + 255) / 256, 256, 0, stream>>>(agg, (long long)NNODES * DIM);
    k_zero<<<(NNODES + 255) / 256, 256, 0, stream>>>(deg, NNODES);
    k_zero<<<(NNODES + 255) / 256, 256, 0, stream>>>(sm, NNODES);
    k_zero<<<(NNODES + 255) / 256, 256, 0, stream>>>((float*)mx, NNODES);

    // pack weights into WMMA fragment order
    k_pack<<<(8192 + 255) / 256, 256, 0, stream>>>(W_mlp,  PmlP,  128, 128, 1);
    k_pack<<<(8192 + 255) / 256, 256, 0, stream>>>(W_umlp, PumlP, 128, 128, 1);
    k_pack<<<(8192 + 255) / 256, 256, 0, stream>>>(W_gat,  PgatP, 128, 128, 0);
    k_pack<<<(4096 + 255) / 256, 256, 0, stream>>>(W_l1,   Pl1,    64, 128, 1);
    k_pack<<<(4096 + 255) / 256, 256, 0, stream>>>(W_g1,   Pg1,    64, 128, 1);

    // x = l2norm(tanh([feat;ufeat] @ W.T + b))   (100000/16 = 6250 tiles)
    k_embed<<<NNODES / 16, 256, 0, stream>>>(features, user_features,
                                             PmlP, b_mlp, PumlP, b_umlp, x);
    // xw = x @ W_gat
    k_gemm_xw<<<NNODES / 16, 256, 0, stream>>>(x, PgatP, xw);

    // GAT edge phase
    k_deg<<<(NEDGE + 255) / 256, 256, 0, stream>>>(src, deg);
    k_edge_logits<<<NEDGE / 8, 256, 0, stream>>>(src, dst, xw, deg, logits, mx);
    k_decode_max<<<(NNODES + 255) / 256, 256, 0, stream>>>(mx, NNODES);
    k_expsum<<<(NEDGE + 255) / 256, 256, 0, stream>>>(dst, logits, (const float*)mx, sm);
    k_scatter<<<NEDGE / 8, 256, 0, stream>>>(src, dst, xw, logits, sm, agg);

    // h = lrelu(l2norm(agg + b_gat))  (in place)
    k_hnorm<<<NNODES / 8, 256, 0, stream>>>(agg, b_gat);

    // rep (fused dual WMMA)
    k_rep<<<NNODES / 16, 128, 0, stream>>>(x, agg, Pl1, b_l1, Pg1, b_g1, id_emb, rep);

    // batch scores + price MLP
    k_batch<<<NBATCH / 8, 256, 0, stream>>>(rep, un, pi, ni, Wp1, bp1, Wp2, bp2,
                                            pos, neg, pred);
}